// CircleLoss_80951543595535
// MI455X (gfx1250) — compile-verified
//
#include <hip/hip_runtime.h>
#include <math.h>

#define N_ROWS 8192
#define DIMS   128
#define MARGIN 0.4f
#define SCALE_F 80.0f
#define NEG_FILL (-1e30f)

typedef __attribute__((ext_vector_type(2))) float v2f;
typedef __attribute__((ext_vector_type(8))) float v8f;

// ---------------------------------------------------------------------------
// Kernel 1: row-wise L2 normalize (8192 blocks x 128 threads) + zero accums.
// ---------------------------------------------------------------------------
__global__ void __launch_bounds__(128)
circle_norm_kernel(const float* __restrict__ x, float* __restrict__ e,
                   float* __restrict__ accum) {
  const int row = blockIdx.x;
  const int tid = threadIdx.x;
  if (row == 0 && tid < 2) accum[tid] = 0.0f;  // loss_sum, valid_cnt

  float v = x[row * DIMS + tid];
  float ss = v * v;
#pragma unroll
  for (int off = 16; off > 0; off >>= 1) ss += __shfl_down(ss, off, 32);

  __shared__ float sred[4];
  const int lane = tid & 31, wv = tid >> 5;
  if (lane == 0) sred[wv] = ss;
  __syncthreads();
  const float tot = sred[0] + sred[1] + sred[2] + sred[3];
  const float nrm = fmaxf(sqrtf(tot), 1e-12f);
  e[row * DIMS + tid] = v / nrm;
}

// ---------------------------------------------------------------------------
// Kernel 2: fused sim-GEMM (fp32 WMMA 16x16x4) + streaming masked logsumexp.
// Grid: 512 blocks of 16 anchors; 256 threads = 8 wave32.
// Each wave sweeps 64 column tiles of 16; sim tile lives only in the WMMA acc.
// B fragments are software-pipelined in two 8-entry register buffers so 8
// global_load_b64 stay in flight while the previous group's WMMAs execute.
// ---------------------------------------------------------------------------
__global__ void __launch_bounds__(256)
circle_main_kernel(const float* __restrict__ E, const int* __restrict__ labels,
                   float* __restrict__ accum) {
  __shared__ float red[8 * 32 * 8 * 6];  // 48 KB of partial LSE states

  const int tid  = threadIdx.x;
  const int lane = tid & 31;
  const int w    = tid >> 5;
  const int anchor_base = blockIdx.x << 4;
  const int lsub = lane & 15;  // N column within tile / M row for A frag
  const int lhi  = lane >> 4;  // half-wave select
  const int mhi  = lhi << 3;   // C/D row offset: 0 or 8

  // Preload all A fragments (K=128 -> 32 x float2 per lane).
  // fp32 A 16x4 layout: lanes 0-15 hold K={0,1}, lanes 16-31 hold K={2,3}.
  const float* Arow = E + (size_t)(anchor_base + lsub) * DIMS + (lhi << 1);
  v2f areg[32];
#pragma unroll
  for (int k = 0; k < 32; ++k) areg[k] = *(const v2f*)(Arow + (k << 2));

  int lblA[8];
#pragma unroll
  for (int r = 0; r < 8; ++r) lblA[r] = labels[anchor_base + mhi + r];

  // Online logsumexp state per (lane, C/D slot): pos & neg {max, sum, count}.
  float pm[8], ps[8], pc[8], nm[8], ns[8], nc[8];
#pragma unroll
  for (int r = 0; r < 8; ++r) {
    pm[r] = NEG_FILL; ps[r] = 0.f; pc[r] = 0.f;
    nm[r] = NEG_FILL; ns[r] = 0.f; nc[r] = 0.f;
  }

  for (int tt = 0; tt < 64; ++tt) {
    const int col_base = (w + (tt << 3)) << 4;
    const int nj   = col_base + lsub;
    const int lblN = labels[nj];
    const float* Brow = E + (size_t)nj * DIMS + (lhi << 1);

    // Double-buffered B fragments: 8 loads in flight per group of 8 WMMAs.
    v2f bbuf[2][8];
#pragma unroll
    for (int k = 0; k < 8; ++k) bbuf[0][k] = *(const v2f*)(Brow + (k << 2));

    v8f acc = {0.f, 0.f, 0.f, 0.f, 0.f, 0.f, 0.f, 0.f};
#pragma unroll
    for (int g = 0; g < 4; ++g) {
      const int cur = g & 1;
      const int nxt = cur ^ 1;
      if (g < 3) {
#pragma unroll
        for (int k = 0; k < 8; ++k)
          bbuf[nxt][k] = *(const v2f*)(Brow + ((((g + 1) << 3) + k) << 2));
      }
#pragma unroll
      for (int k = 0; k < 8; ++k) {
        acc = __builtin_amdgcn_wmma_f32_16x16x4_f32(
            /*neg_a=*/false, areg[(g << 3) + k], /*neg_b=*/false, bbuf[cur][k],
            /*c_mod=*/(short)0, acc, /*reuse_a=*/false, /*reuse_b=*/false);
      }
    }

    // Epilogue on the 16x16 tile held in acc (lane slot r -> M=mhi+r, N=lsub).
#pragma unroll
    for (int r = 0; r < 8; ++r) {
      const float s  = acc[r];
      const int   i  = anchor_base + mhi + r;
      const bool same = (lblA[r] == lblN);
      const bool isp  = same && (i != nj);
      const bool isn  = !same;
      const float ap = -SCALE_F * fmaxf(1.0f + MARGIN - s, 0.f) * (s - 1.0f + MARGIN);
      const float an =  SCALE_F * fmaxf(s + MARGIN, 0.f) * (s - MARGIN);
      const float tp = isp ? ap : NEG_FILL;
      const float tn = isn ? an : NEG_FILL;
      float mm = fmaxf(pm[r], tp);
      ps[r] = ps[r] * __expf(pm[r] - mm) + __expf(tp - mm);
      pm[r] = mm;
      mm = fmaxf(nm[r], tn);
      ns[r] = ns[r] * __expf(nm[r] - mm) + __expf(tn - mm);
      nm[r] = mm;
      pc[r] += isp ? 1.f : 0.f;
      nc[r] += isn ? 1.f : 0.f;
    }
  }

  // Spill per-lane partial states to LDS.
#pragma unroll
  for (int r = 0; r < 8; ++r) {
    float* p = &red[((((w << 5) | lane) << 3) + r) * 6];
    p[0] = pm[r]; p[1] = ps[r]; p[2] = pc[r];
    p[3] = nm[r]; p[4] = ns[r]; p[5] = nc[r];
  }
  __syncthreads();

  // Threads 0..15 each finish one anchor row: merge 8 waves x 16 lane-columns.
  if (tid < 16) {
    const int m    = tid;
    const int hi16 = (m >> 3) << 4;  // lanes 0-15 hold rows 0-7; 16-31 rows 8-15
    const int r    = m & 7;
    float PM = NEG_FILL, PS = 0.f, PC = 0.f;
    float NM = NEG_FILL, NS = 0.f, NC = 0.f;
    for (int wv = 0; wv < 8; ++wv) {
      for (int ln = 0; ln < 16; ++ln) {
        const float* p = &red[((((wv << 5) | (ln + hi16)) << 3) + r) * 6];
        float m2 = p[0], s2 = p[1];
        float mm = fmaxf(PM, m2);
        PS = PS * __expf(PM - mm) + s2 * __expf(m2 - mm);
        PM = mm; PC += p[2];
        m2 = p[3]; s2 = p[4];
        mm = fmaxf(NM, m2);
        NS = NS * __expf(NM - mm) + s2 * __expf(m2 - mm);
        NM = mm; NC += p[5];
      }
    }
    if (PC > 0.5f && NC > 0.5f) {
      const float lse_p = PM + __logf(PS);
      const float lse_n = NM + __logf(NS);
      const float z = lse_p + __logf(NC) + lse_n + __logf(PC);
      const float li = fmaxf(z, 0.f) + log1pf(__expf(-fabsf(z)));  // softplus
      atomicAdd(&accum[0], li);
      atomicAdd(&accum[1], 1.0f);
    }
  }
}

// ---------------------------------------------------------------------------
// Kernel 3: final mean.
// ---------------------------------------------------------------------------
__global__ void circle_final_kernel(const float* __restrict__ accum,
                                    float* __restrict__ out) {
  out[0] = accum[0] / fmaxf(accum[1], 1.0f);
}

extern "C" void kernel_launch(void* const* d_in, const int* in_sizes, int n_in,
                              void* d_out, int out_size, void* d_ws, size_t ws_size,
                              hipStream_t stream) {
  const float* embeds = (const float*)d_in[0];
  const int*   labels = (const int*)d_in[1];  // jnp.int64 -> int32 under default JAX x64=off
  float* out = (float*)d_out;

  float* E     = (float*)d_ws;               // 8192*128 normalized embeds (4 MB)
  float* accum = E + (size_t)N_ROWS * DIMS;  // [loss_sum, valid_cnt]

  circle_norm_kernel<<<N_ROWS, DIMS, 0, stream>>>(embeds, E, accum);
  circle_main_kernel<<<N_ROWS / 16, 256, 0, stream>>>(E, labels, accum);
  circle_final_kernel<<<1, 1, 0, stream>>>(accum, out);
}